// GpuNUFFTSingleCoil_32074815766959
// MI455X (gfx1250) — compile-verified
//
#include <hip/hip_runtime.h>
#include <hip/hip_bf16.h>

// ---------------------------------------------------------------------------
// NUFFT forward (gpuNUFFT single coil) for gfx1250.
// Strategy: centered-DFT-as-GEMM on the f32 WMMA path.
//   kgrid = (1/G) * Fsub * img * Fsub^T   (Fsub = centered DFT cols pad..pad+N)
// PREP : X = (x_re + i x_im) * ci[k] * ci[n]      (deapodized image)
// GEMM1: T = Fsub(640x320) * X(320x320)           (32x32 tile / wave)
// GEMM2: Ggrid = T(640x320) * Ft(320x640) / G     (32x32 tile / wave)
// INTERP: 3x3 Kaiser-Bessel gather at 200k pts x 8 batches, * sqrt(dcf)
// ---------------------------------------------------------------------------

#define GDIM 640
#define NDIM 320
#define PAD  160
#define NB   8
#define NM   200000
#define BETA 6.4860763f     // pi*sqrt((W/OSF)^2*(OSF-0.5)^2 - 0.8), W=3, OSF=2
#define PI_F 3.14159265358979323846f

typedef __attribute__((ext_vector_type(2))) float v2f;
typedef __attribute__((ext_vector_type(8))) float v8f;

static __device__ __forceinline__ v8f wmma_f32(v2f a, v2f b, v8f c) {
  // V_WMMA_F32_16X16X4_F32 : D(16x16 f32) = A(16x4) * B(4x16) + C
  return __builtin_amdgcn_wmma_f32_16x16x4_f32(
      /*neg_a=*/false, a, /*neg_b=*/false, b,
      /*c_mod=*/(short)0, c, /*reuse_a=*/false, /*reuse_b=*/false);
}

// ---------------------------------------------------------------------------
// Workspace layout (floats)
// ---------------------------------------------------------------------------
#define OFF_CI   ((size_t)0)                          // 320 (reserve 512)
#define OFF_FRE  ((size_t)512)                        // 640*320
#define OFF_FIM  (OFF_FRE + (size_t)GDIM * NDIM)
#define OFF_FTRE (OFF_FIM + (size_t)GDIM * NDIM)      // 320*640 (transposed)
#define OFF_FTIM (OFF_FTRE + (size_t)GDIM * NDIM)
#define OFF_TRE  (OFF_FTIM + (size_t)GDIM * NDIM)     // B*640*320
#define OFF_TIM  (OFF_TRE + (size_t)NB * GDIM * NDIM)
#define OFF_GRE  (OFF_TIM + (size_t)NB * GDIM * NDIM) // B*640*640
#define OFF_GIM  (OFF_GRE + (size_t)NB * GDIM * GDIM)
#define OFF_XRE  (OFF_GIM + (size_t)NB * GDIM * GDIM) // B*320*320 (deapodized)
#define OFF_XIM  (OFF_XRE + (size_t)NB * NDIM * NDIM)

// ---------------------------------------------------------------------------
// Init: 1/deapodization LUT
// ---------------------------------------------------------------------------
__global__ void init_deapod_kernel(float* __restrict__ ci) {
  int i = threadIdx.x;
  if (i >= NDIM) return;
  float xs  = (float)(i - NDIM / 2) / (float)GDIM;
  float pwx = PI_F * 3.0f * xs;
  float arg = BETA * BETA - pwx * pwx;
  float s   = sqrtf(fabsf(arg));
  float val;
  if (arg > 0.0f) {
    val = sinhf(s) / s;
  } else {
    val = (s < 1e-8f) ? 1.0f : (sinf(s) / s);
  }
  ci[i] = 1.0f / val;
}

// ---------------------------------------------------------------------------
// Init: centered DFT matrix slice Fsub[m][k] = exp(-2*pi*i*(m-320)*(k-160)/640)
// plus its transpose for coalesced B loads in GEMM2. Integer phase reduction
// mod G keeps __sincosf accurate.
// ---------------------------------------------------------------------------
__global__ void init_dft_kernel(float* __restrict__ Fre, float* __restrict__ Fim,
                                float* __restrict__ Ftre, float* __restrict__ Ftim) {
  int t = blockIdx.x * blockDim.x + threadIdx.x;
  if (t >= GDIM * NDIM) return;
  int m = t / NDIM;
  int k = t - m * NDIM;
  int prod = (m - GDIM / 2) * (k + PAD - GDIM / 2);
  int r = prod % GDIM;                       // in (-G, G)
  float ph = (-2.0f * PI_F / (float)GDIM) * (float)r;
  float s, c;
  __sincosf(ph, &s, &c);
  Fre[m * NDIM + k]  = c;
  Fim[m * NDIM + k]  = s;
  Ftre[k * GDIM + m] = c;
  Ftim[k * GDIM + m] = s;
}

// ---------------------------------------------------------------------------
// Prep: deapodized image X = (x_re + i x_im) * ci[k] * ci[n]
// (hoists the per-element scaling out of GEMM1's inner loop)
// ---------------------------------------------------------------------------
__global__ __launch_bounds__(256) void deapod_img_kernel(
    const float* __restrict__ xre, const float* __restrict__ xim,
    const float* __restrict__ ci,
    float* __restrict__ Xre, float* __restrict__ Xim) {
  int t = blockIdx.x * blockDim.x + threadIdx.x;
  if (t >= NB * NDIM * NDIM) return;
  int rem = t % (NDIM * NDIM);
  int k = rem / NDIM;
  int n = rem - k * NDIM;
  float s = ci[k] * ci[n];
  Xre[t] = xre[t] * s;
  Xim[t] = xim[t] * s;
}

// ---------------------------------------------------------------------------
// GEMM1: T(640x320) = Fsub(640x320) * X(320x320), complex.
// One wave per 32x32 output tile: 2 M-tiles x 2 N-tiles, 16 WMMAs / K=4 step.
// ---------------------------------------------------------------------------
__global__ __launch_bounds__(128) void gemm1_kernel(
    const float* __restrict__ Xre, const float* __restrict__ Xim,
    const float* __restrict__ Fre, const float* __restrict__ Fim,
    float* __restrict__ Tre, float* __restrict__ Tim) {
  const int TM2 = GDIM / 32;     // 20
  const int TN2 = NDIM / 32;     // 10
  const int PERB = TM2 * TN2;    // 200
  int wid = blockIdx.x * (blockDim.x >> 5) + (threadIdx.x >> 5);
  if (wid >= NB * PERB) return;  // wave-uniform guard

  int b   = wid / PERB;
  int rem = wid - b * PERB;
  int tm2 = rem / TN2;
  int tn2 = rem - tm2 * TN2;

  int l    = threadIdx.x & 31;
  int half = l >> 4;
  int lr   = l & 15;

  int rowbase = tm2 * 32;
  int colbase = tn2 * 32;
  int row0 = rowbase + lr;
  int row1 = row0 + 16;
  int col0 = colbase + lr;
  int col1 = col0 + 16;

  const float* xr = Xre + (size_t)b * NDIM * NDIM;
  const float* xi = Xim + (size_t)b * NDIM * NDIM;

  v8f are00 = {}, aim00 = {}, are01 = {}, aim01 = {};
  v8f are10 = {}, aim10 = {}, are11 = {}, aim11 = {};

  for (int k0 = 0; k0 < NDIM; k0 += 4) {
    int ka = k0 + (half << 1);    // this lane's K base

    float2 f0r = *(const float2*)(Fre + (size_t)row0 * NDIM + ka);
    float2 f0i = *(const float2*)(Fim + (size_t)row0 * NDIM + ka);
    float2 f1r = *(const float2*)(Fre + (size_t)row1 * NDIM + ka);
    float2 f1i = *(const float2*)(Fim + (size_t)row1 * NDIM + ka);
    v2f a0re = {f0r.x, f0r.y};
    v2f a0im = {f0i.x, f0i.y};
    v2f a1re = {f1r.x, f1r.y};
    v2f a1im = {f1i.x, f1i.y};
    v2f a0imn = -a0im;            // f32 WMMA has no A/B NEG; negate in VALU
    v2f a1imn = -a1im;

    const float* xr0 = xr + (size_t)ka * NDIM;
    const float* xi0 = xi + (size_t)ka * NDIM;
    v2f br0 = {xr0[col0], xr0[NDIM + col0]};
    v2f bi0 = {xi0[col0], xi0[NDIM + col0]};
    v2f br1 = {xr0[col1], xr0[NDIM + col1]};
    v2f bi1 = {xi0[col1], xi0[NDIM + col1]};

    are00 = wmma_f32(a0re, br0, are00);
    are00 = wmma_f32(a0imn, bi0, are00);
    aim00 = wmma_f32(a0re, bi0, aim00);
    aim00 = wmma_f32(a0im, br0, aim00);
    are01 = wmma_f32(a0re, br1, are01);
    are01 = wmma_f32(a0imn, bi1, are01);
    aim01 = wmma_f32(a0re, bi1, aim01);
    aim01 = wmma_f32(a0im, br1, aim01);
    are10 = wmma_f32(a1re, br0, are10);
    are10 = wmma_f32(a1imn, bi0, are10);
    aim10 = wmma_f32(a1re, bi0, aim10);
    aim10 = wmma_f32(a1im, br0, aim10);
    are11 = wmma_f32(a1re, br1, are11);
    are11 = wmma_f32(a1imn, bi1, are11);
    aim11 = wmma_f32(a1re, bi1, aim11);
    aim11 = wmma_f32(a1im, br1, aim11);
  }

  float* tr = Tre + (size_t)b * GDIM * NDIM;
  float* ti = Tim + (size_t)b * GDIM * NDIM;
#pragma unroll
  for (int r = 0; r < 8; ++r) {
    int o0 = rowbase + r + (half << 3);
    int o1 = o0 + 16;
    tr[(size_t)o0 * NDIM + col0] = are00[r];
    ti[(size_t)o0 * NDIM + col0] = aim00[r];
    tr[(size_t)o0 * NDIM + col1] = are01[r];
    ti[(size_t)o0 * NDIM + col1] = aim01[r];
    tr[(size_t)o1 * NDIM + col0] = are10[r];
    ti[(size_t)o1 * NDIM + col0] = aim10[r];
    tr[(size_t)o1 * NDIM + col1] = are11[r];
    ti[(size_t)o1 * NDIM + col1] = aim11[r];
  }
}

// ---------------------------------------------------------------------------
// GEMM2: Ggrid(640x640) = T(640x320) * Ft(320x640) / G, complex.
// One wave per 32x32 output tile.
// ---------------------------------------------------------------------------
__global__ __launch_bounds__(128) void gemm2_kernel(
    const float* __restrict__ Tre, const float* __restrict__ Tim,
    const float* __restrict__ Ftre, const float* __restrict__ Ftim,
    float* __restrict__ Gre, float* __restrict__ Gim) {
  const int TM2 = GDIM / 32;     // 20
  const int TN2 = GDIM / 32;     // 20
  const int PERB = TM2 * TN2;    // 400
  int wid = blockIdx.x * (blockDim.x >> 5) + (threadIdx.x >> 5);
  if (wid >= NB * PERB) return;

  int b   = wid / PERB;
  int rem = wid - b * PERB;
  int tm2 = rem / TN2;
  int tn2 = rem - tm2 * TN2;

  int l    = threadIdx.x & 31;
  int half = l >> 4;
  int lr   = l & 15;

  int rowbase = tm2 * 32;
  int colbase = tn2 * 32;
  int row0 = rowbase + lr;
  int row1 = row0 + 16;
  int col0 = colbase + lr;
  int col1 = col0 + 16;

  const float* tr = Tre + (size_t)b * GDIM * NDIM;
  const float* ti = Tim + (size_t)b * GDIM * NDIM;

  v8f are00 = {}, aim00 = {}, are01 = {}, aim01 = {};
  v8f are10 = {}, aim10 = {}, are11 = {}, aim11 = {};

  for (int k0 = 0; k0 < NDIM; k0 += 4) {
    int ka = k0 + (half << 1);

    float2 a0r = *(const float2*)(tr + (size_t)row0 * NDIM + ka);
    float2 a0i = *(const float2*)(ti + (size_t)row0 * NDIM + ka);
    float2 a1r = *(const float2*)(tr + (size_t)row1 * NDIM + ka);
    float2 a1i = *(const float2*)(ti + (size_t)row1 * NDIM + ka);
    v2f a0re = {a0r.x, a0r.y};
    v2f a0im = {a0i.x, a0i.y};
    v2f a1re = {a1r.x, a1r.y};
    v2f a1im = {a1i.x, a1i.y};
    v2f a0imn = -a0im;
    v2f a1imn = -a1im;

    const float* fr0 = Ftre + (size_t)ka * GDIM;
    const float* fi0 = Ftim + (size_t)ka * GDIM;
    v2f br0 = {fr0[col0], fr0[GDIM + col0]};
    v2f bi0 = {fi0[col0], fi0[GDIM + col0]};
    v2f br1 = {fr0[col1], fr0[GDIM + col1]};
    v2f bi1 = {fi0[col1], fi0[GDIM + col1]};

    are00 = wmma_f32(a0re, br0, are00);
    are00 = wmma_f32(a0imn, bi0, are00);
    aim00 = wmma_f32(a0re, bi0, aim00);
    aim00 = wmma_f32(a0im, br0, aim00);
    are01 = wmma_f32(a0re, br1, are01);
    are01 = wmma_f32(a0imn, bi1, are01);
    aim01 = wmma_f32(a0re, bi1, aim01);
    aim01 = wmma_f32(a0im, br1, aim01);
    are10 = wmma_f32(a1re, br0, are10);
    are10 = wmma_f32(a1imn, bi0, are10);
    aim10 = wmma_f32(a1re, bi0, aim10);
    aim10 = wmma_f32(a1im, br0, aim10);
    are11 = wmma_f32(a1re, br1, are11);
    are11 = wmma_f32(a1imn, bi1, are11);
    aim11 = wmma_f32(a1re, bi1, aim11);
    aim11 = wmma_f32(a1im, br1, aim11);
  }

  const float inv = 1.0f / (float)GDIM;
  float* gr = Gre + (size_t)b * GDIM * GDIM;
  float* gi = Gim + (size_t)b * GDIM * GDIM;
#pragma unroll
  for (int r = 0; r < 8; ++r) {
    int o0 = rowbase + r + (half << 3);
    int o1 = o0 + 16;
    gr[(size_t)o0 * GDIM + col0] = are00[r] * inv;
    gi[(size_t)o0 * GDIM + col0] = aim00[r] * inv;
    gr[(size_t)o0 * GDIM + col1] = are01[r] * inv;
    gi[(size_t)o0 * GDIM + col1] = aim01[r] * inv;
    gr[(size_t)o1 * GDIM + col0] = are10[r] * inv;
    gi[(size_t)o1 * GDIM + col0] = aim10[r] * inv;
    gr[(size_t)o1 * GDIM + col1] = are11[r] * inv;
    gi[(size_t)o1 * GDIM + col1] = aim11[r] * inv;
  }
}

// ---------------------------------------------------------------------------
// Interpolation: 3x3 Kaiser-Bessel gather, * sqrt(dcf).
// ---------------------------------------------------------------------------
static __device__ __forceinline__ float i0f(float x) {
  // Abramowitz & Stegun 9.8.1 / 9.8.2 (x >= 0 here)
  float ax = fabsf(x);
  if (ax < 3.75f) {
    float t = ax / 3.75f;
    t *= t;
    return 1.0f + t * (3.5156229f + t * (3.0899424f + t * (1.2067492f +
           t * (0.2659732f + t * (0.0360768f + t * 0.0045813f)))));
  }
  float t = 3.75f / ax;
  float p = 0.39894228f + t * (0.01328592f + t * (0.00225319f +
            t * (-0.00157565f + t * (0.00916281f + t * (-0.02057706f +
            t * (0.02635537f + t * (-0.01647633f + t * 0.00392377f)))))));
  return expf(ax) * rsqrtf(ax) * p;
}

static __device__ __forceinline__ float kbw(float d) {
  float ad = fabsf(d);
  float u  = d * (2.0f / 3.0f);               // 2d/W, W=3
  float tt = fmaxf(1.0f - u * u, 0.0f);
  float v  = i0f(BETA * sqrtf(tt));
  return (ad <= 1.5f) ? v : 0.0f;
}

static __device__ __forceinline__ int wrapG(int v) {
  int r = v % GDIM;
  return (r < 0) ? r + GDIM : r;
}

__global__ __launch_bounds__(256) void interp_kernel(
    const float* __restrict__ traj, const float* __restrict__ dcf,
    const float* __restrict__ Gre, const float* __restrict__ Gim,
    float* __restrict__ out) {
  int t = blockIdx.x * blockDim.x + threadIdx.x;
  if (t >= NB * NM) return;
  int b = t / NM;
  int m = t - b * NM;

  float tx = traj[((size_t)b * 2 + 0) * NM + m];
  float ty = traj[((size_t)b * 2 + 1) * NM + m];
  float cx = (tx + 0.5f) * (float)GDIM;
  float cy = (ty + 0.5f) * (float)GDIM;
  int bx = (int)floorf(cx - 1.0f);
  int by = (int)floorf(cy - 1.0f);

  float wx[3], wy[3];
  int ix[3], iy[3];
#pragma unroll
  for (int k = 0; k < 3; ++k) {
    ix[k] = wrapG(bx + k);
    iy[k] = wrapG(by + k);
    wx[k] = kbw(cx - (float)(bx + k));
    wy[k] = kbw(cy - (float)(by + k));
  }

  const float* gr = Gre + (size_t)b * GDIM * GDIM;
  const float* gi = Gim + (size_t)b * GDIM * GDIM;

  float sr = 0.0f, si = 0.0f;
#pragma unroll
  for (int a = 0; a < 3; ++a) {
    size_t rowoff = (size_t)iy[a] * GDIM;
    float wya = wy[a];
#pragma unroll
    for (int c = 0; c < 3; ++c) {
      float w = wya * wx[c];
      size_t idx = rowoff + (size_t)ix[c];
      sr += w * gr[idx];
      si += w * gi[idx];
    }
  }

  float sd = sqrtf(dcf[(size_t)b * NM + m]);
  out[(size_t)t * 2 + 0] = sr * sd;
  out[(size_t)t * 2 + 1] = si * sd;
}

// ---------------------------------------------------------------------------
// Launch
// ---------------------------------------------------------------------------
extern "C" void kernel_launch(void* const* d_in, const int* in_sizes, int n_in,
                              void* d_out, int out_size, void* d_ws, size_t ws_size,
                              hipStream_t stream) {
  const float* xre  = (const float*)d_in[0];
  const float* xim  = (const float*)d_in[1];
  const float* traj = (const float*)d_in[2];
  const float* dcf  = (const float*)d_in[3];
  float* out = (float*)d_out;
  float* ws  = (float*)d_ws;

  float* ci   = ws + OFF_CI;
  float* Fre  = ws + OFF_FRE;
  float* Fim  = ws + OFF_FIM;
  float* Ftre = ws + OFF_FTRE;
  float* Ftim = ws + OFF_FTIM;
  float* Tre  = ws + OFF_TRE;
  float* Tim  = ws + OFF_TIM;
  float* Gre  = ws + OFF_GRE;
  float* Gim  = ws + OFF_GIM;
  float* Xre  = ws + OFF_XRE;
  float* Xim  = ws + OFF_XIM;

  // 1) deapodization LUT
  init_deapod_kernel<<<1, NDIM, 0, stream>>>(ci);

  // 2) centered DFT matrix (+transpose)
  {
    int total = GDIM * NDIM;
    init_dft_kernel<<<(total + 255) / 256, 256, 0, stream>>>(Fre, Fim, Ftre, Ftim);
  }

  // 3) deapodized image
  {
    int total = NB * NDIM * NDIM;
    deapod_img_kernel<<<(total + 255) / 256, 256, 0, stream>>>(xre, xim, ci, Xre, Xim);
  }

  // 4) GEMM1: T = Fsub * X  (8 * 20 * 10 = 1600 wave-tiles, 4 waves/block)
  gemm1_kernel<<<(NB * 20 * 10) / 4, 128, 0, stream>>>(Xre, Xim, Fre, Fim, Tre, Tim);

  // 5) GEMM2: Ggrid = T * Ft / G  (8 * 20 * 20 = 3200 wave-tiles)
  gemm2_kernel<<<(NB * 20 * 20) / 4, 128, 0, stream>>>(Tre, Tim, Ftre, Ftim, Gre, Gim);

  // 6) interpolation gather
  {
    int total = NB * NM;
    interp_kernel<<<(total + 255) / 256, 256, 0, stream>>>(traj, dcf, Gre, Gim, out);
  }
}